// LightGCN_6803228197244
// MI455X (gfx1250) — compile-verified
//
#include <hip/hip_runtime.h>
#include <hip/hip_bf16.h>

#define NUM_USERS 150000
#define NUM_ITEMS 75000
#define NUM_NODES (NUM_USERS + NUM_ITEMS)
#define EMB_DIM   64
#define NUM_LAYERS 3
#define NUM_EDGES 5000000
#define BATCH     8192
#define LAYER_W   0.25f

typedef float v2f __attribute__((ext_vector_type(2)));
typedef float v8f __attribute__((ext_vector_type(8)));

// ---------------------------------------------------------------------------
// Kernel 1: all_emb = concat(user_table, item_table); light_out = w*all_emb;
//           acc = 0.   Vectorized float4 (a float4 never straddles tables,
//           rows are 16 float4s).
// ---------------------------------------------------------------------------
__global__ void lgcn_init(const float* __restrict__ user_table,
                          const float* __restrict__ item_table,
                          float* __restrict__ A,          // all_emb (ws)
                          float* __restrict__ acc,        // segment-sum accumulator (ws)
                          float* __restrict__ light_out)  // d_out + 2*BATCH
{
    const int n4 = NUM_NODES * (EMB_DIM / 4);
    int i4 = blockIdx.x * blockDim.x + threadIdx.x;
    if (i4 >= n4) return;
    const int off = i4 * 4;
    float4 v;
    if (off < NUM_USERS * EMB_DIM)
        v = *reinterpret_cast<const float4*>(user_table + off);
    else
        v = *reinterpret_cast<const float4*>(item_table + (off - NUM_USERS * EMB_DIM));
    *reinterpret_cast<float4*>(A + off) = v;
    float4 lo = make_float4(LAYER_W * v.x, LAYER_W * v.y, LAYER_W * v.z, LAYER_W * v.w);
    *reinterpret_cast<float4*>(light_out + off) = lo;
    *reinterpret_cast<float4*>(acc + off) = make_float4(0.f, 0.f, 0.f, 0.f);
}

// ---------------------------------------------------------------------------
// Kernel 2: SpMM scatter:  acc[rows[e]] += vals[e] * A[cols[e]]
// 16 lanes per edge, float4 gather (hits L2: table is 57.6MB of 192MB),
// non-temporal loads for the 60MB edge stream so it does not evict the
// L2-resident tables, plus prefetch-ahead (global_prefetch_b8).
// f32 global atomics into the L2-resident accumulator.
// ---------------------------------------------------------------------------
__global__ void lgcn_spmm_scatter(const int*   __restrict__ rows,
                                  const int*   __restrict__ cols,
                                  const float* __restrict__ vals,
                                  const float* __restrict__ A,
                                  float*       __restrict__ acc)
{
    const long long t = (long long)blockIdx.x * blockDim.x + threadIdx.x;
    const long long e = t >> 4;              // 16 threads per edge
    if (e >= NUM_EDGES) return;
    const int sub = (int)(t & 15);

    if (sub == 0 && e + 4096 < NUM_EDGES) {  // stream-ahead hints
        __builtin_prefetch(cols + e + 4096, 0, 0);
        __builtin_prefetch(rows + e + 4096, 0, 0);
        __builtin_prefetch(vals + e + 4096, 0, 0);
    }

    const int   c = __builtin_nontemporal_load(cols + e);
    const int   r = __builtin_nontemporal_load(rows + e);
    const float v = __builtin_nontemporal_load(vals + e);

    const float4 g = *reinterpret_cast<const float4*>(A + (size_t)c * EMB_DIM + sub * 4);
    float* dst = acc + (size_t)r * EMB_DIM + sub * 4;
    unsafeAtomicAdd(dst + 0, g.x * v);   // global_atomic_add_f32
    unsafeAtomicAdd(dst + 1, g.y * v);
    unsafeAtomicAdd(dst + 2, g.z * v);
    unsafeAtomicAdd(dst + 3, g.w * v);
}

// ---------------------------------------------------------------------------
// Kernel 3: per-row L2 normalize + light_out += w*x + rezero acc.
// One wave32 per row, float2 per lane, shfl_xor tree reduction.
// ---------------------------------------------------------------------------
__global__ void lgcn_normalize_accum(float* __restrict__ acc,
                                     float* __restrict__ A,
                                     float* __restrict__ light_out)
{
    const int row  = (blockIdx.x * blockDim.x + threadIdx.x) >> 5;
    const int lane = threadIdx.x & 31;
    if (row >= NUM_NODES) return;
    const size_t off = (size_t)row * EMB_DIM + lane * 2;

    float2 x = *reinterpret_cast<float2*>(acc + off);
    float ss = x.x * x.x + x.y * x.y;
    #pragma unroll
    for (int o = 16; o > 0; o >>= 1) ss += __shfl_xor(ss, o, 32);
    const float rn = rsqrtf(fmaxf(ss, 1e-12f));

    float2 y = make_float2(x.x * rn, x.y * rn);
    *reinterpret_cast<float2*>(A + off) = y;              // input for next layer
    float2 lo = *reinterpret_cast<float2*>(light_out + off);
    lo.x += LAYER_W * y.x;
    lo.y += LAYER_W * y.y;
    *reinterpret_cast<float2*>(light_out + off) = lo;
    *reinterpret_cast<float2*>(acc + off) = make_float2(0.f, 0.f); // rezero (deterministic replay)
}

// ---------------------------------------------------------------------------
// Kernel 4: BPR scoring via V_WMMA_F32_16X16X4_F32.
// Each wave handles 16 (user, pos, neg) triples:
//   D = U(16x4) * P^T(4x16) accumulated over 16 K-steps (K=64 total);
//   diag(D) = the 16 dot products, full f32 precision.
// A layout (16x4 f32): lane l -> M = l&15, K = (l>>4)*2 (+0 in VGPR0, +1 in VGPR1).
// B layout (4x16 f32): lane l -> N = l&15, same K striping.
// D layout: D[r][r] in VGPR r @ lane r; D[r+8][r+8] in VGPR r @ lane r+24.
// 512 full waves, EXEC all-1s, no divergence before the WMMAs.
// ---------------------------------------------------------------------------
__global__ void lgcn_score_wmma(const int*   __restrict__ users,
                                const int*   __restrict__ pos_items,
                                const int*   __restrict__ neg_items,
                                const float* __restrict__ light_out,
                                float*       __restrict__ pos_out,
                                float*       __restrict__ neg_out)
{
    const int wave = (blockIdx.x * blockDim.x + threadIdx.x) >> 5;
    const int lane = threadIdx.x & 31;
    const int t = wave * 16;
    if (t >= BATCH) return;

    const int m    = lane & 15;
    const int koff = (lane >> 4) * 2;

    const float* urow = light_out + (size_t)users[t + m] * EMB_DIM;
    const float* prow = light_out + ((size_t)NUM_USERS + pos_items[t + m]) * EMB_DIM;
    const float* nrow = light_out + ((size_t)NUM_USERS + neg_items[t + m]) * EMB_DIM;

    v8f accP = {0.f, 0.f, 0.f, 0.f, 0.f, 0.f, 0.f, 0.f};
    v8f accN = {0.f, 0.f, 0.f, 0.f, 0.f, 0.f, 0.f, 0.f};

    #pragma unroll
    for (int s = 0; s < 16; ++s) {
        const int k0 = 4 * s + koff;
        v2f a;  a.x  = urow[k0]; a.y  = urow[k0 + 1];
        v2f bp; bp.x = prow[k0]; bp.y = prow[k0 + 1];
        v2f bn; bn.x = nrow[k0]; bn.y = nrow[k0 + 1];
        accP = __builtin_amdgcn_wmma_f32_16x16x4_f32(false, a, false, bp,
                                                     (short)0, accP, false, false);
        accN = __builtin_amdgcn_wmma_f32_16x16x4_f32(false, a, false, bn,
                                                     (short)0, accN, false, false);
    }

    #pragma unroll
    for (int r = 0; r < 8; ++r) {
        if (lane == r) {
            pos_out[t + r] = accP[r];
            neg_out[t + r] = accN[r];
        }
        if (lane == r + 24) {
            pos_out[t + r + 8] = accP[r];
            neg_out[t + r + 8] = accN[r];
        }
    }
}

// ---------------------------------------------------------------------------
// Launch
// ---------------------------------------------------------------------------
extern "C" void kernel_launch(void* const* d_in, const int* in_sizes, int n_in,
                              void* d_out, int out_size, void* d_ws, size_t ws_size,
                              hipStream_t stream) {
    (void)in_sizes; (void)n_in; (void)out_size; (void)ws_size;

    const int*   users      = (const int*)  d_in[0];
    const int*   pos_items  = (const int*)  d_in[1];
    const int*   neg_items  = (const int*)  d_in[2];
    const int*   adj_rows   = (const int*)  d_in[3];
    const int*   adj_cols   = (const int*)  d_in[4];
    const float* adj_vals   = (const float*)d_in[5];
    const float* user_table = (const float*)d_in[6];
    const float* item_table = (const float*)d_in[7];

    float* out       = (float*)d_out;
    float* pos_out   = out;                 // [0, 8192)
    float* neg_out   = out + BATCH;         // [8192, 16384)
    float* light_out = out + 2 * BATCH;     // user_emb ++ item_emb, 225000 x 64 contiguous

    float* A   = (float*)d_ws;                            // 57.6 MB
    float* acc = A + (size_t)NUM_NODES * EMB_DIM;         // 57.6 MB

    // init
    {
        const int n4 = NUM_NODES * (EMB_DIM / 4);
        lgcn_init<<<(n4 + 255) / 256, 256, 0, stream>>>(
            user_table, item_table, A, acc, light_out);
    }

    // 3 LightGCN layers: scatter-add, then normalize+accumulate (rezeros acc)
    for (int layer = 0; layer < NUM_LAYERS; ++layer) {
        const long long spmm_threads = (long long)NUM_EDGES * 16;
        lgcn_spmm_scatter<<<(unsigned)((spmm_threads + 255) / 256), 256, 0, stream>>>(
            adj_rows, adj_cols, adj_vals, A, acc);

        const long long norm_threads = (long long)NUM_NODES * 32;
        lgcn_normalize_accum<<<(unsigned)((norm_threads + 255) / 256), 256, 0, stream>>>(
            acc, A, light_out);
    }

    // WMMA scoring: 512 waves of 16 pairs each
    {
        const int score_threads = (BATCH / 16) * 32;
        lgcn_score_wmma<<<score_threads / 256, 256, 0, stream>>>(
            users, pos_items, neg_items, light_out, pos_out, neg_out);
    }
}